// Sparsemax_1812476199363
// MI455X (gfx1250) — compile-verified
//
#include <hip/hip_runtime.h>
#include <cstdint>
#include <cstddef>

// Sparsemax, row-wise over [4096, 32000] fp32.
// One workgroup per row; whole row (125 KB) staged into LDS via CDNA5 async
// global->LDS copies (ASYNCcnt path). tau found by Michelot fixed-point
// iteration (exact simplex projection, no sort); once the active set fits in
// CAP, survivors are compacted (deterministically, no atomics) into a small
// LDS buffer so the remaining iterations touch ~16-31 KB instead of 128 KB.
// Single HBM read pass + single HBM write pass.

#define ROW_D   32000
#define ROW_V4  (ROW_D / 4)     // 8000 float4 per row
#define BLOCK   512             // 16 wave32 waves
#define NWAVE   (BLOCK / 32)
#define ITERS   16
#define CAP     7936            // candidate buffer (floats); LDS total = 160000 B

// Async global->LDS builtin takes int4 pointers (AS1 src, AS3 dst).
typedef int v4i __attribute__((vector_size(16)));
typedef __attribute__((address_space(1))) v4i* g_v4i_p;
typedef __attribute__((address_space(3))) v4i* l_v4i_p;

__global__ __launch_bounds__(BLOCK) void sparsemax_row_kernel(
    const float* __restrict__ x, float* __restrict__ out) {
  extern __shared__ float smem[];
  float* row   = smem;                   // ROW_D floats
  float* cand  = smem + ROW_D;           // CAP floats (compacted active set)
  float* redA  = cand + CAP;             // NWAVE float partials
  float* redB  = redA + NWAVE;           // NWAVE float partials
  int*   iredA = (int*)(redB + NWAVE);   // NWAVE int partials (scan)
  int*   iredB = iredA + NWAVE;          // NWAVE int partials (scan)

  const int tid  = threadIdx.x;
  const int lane = tid & 31;
  const int wave = tid >> 5;

  const float* __restrict__ xr   = x   + (size_t)blockIdx.x * ROW_D;
  float*       __restrict__ outr = out + (size_t)blockIdx.x * ROW_D;

  // ---------------- Stage row into LDS (async global -> LDS, b128) ---------
#if __has_builtin(__builtin_amdgcn_global_load_async_to_lds_b128)
  for (int i = tid; i < ROW_V4; i += BLOCK) {
    __builtin_amdgcn_global_load_async_to_lds_b128(
        (g_v4i_p)(xr + 4 * i), (l_v4i_p)(row + 4 * i),
        /*offset=*/0, /*cpol=*/0);
  }
#if __has_builtin(__builtin_amdgcn_s_wait_asynccnt)
  __builtin_amdgcn_s_wait_asynccnt(0);
#else
  asm volatile("s_wait_asynccnt 0" ::: "memory");
#endif
#else
  {
    const float4* xr4  = (const float4*)xr;
    float4*       row4 = (float4*)row;
    for (int i = tid; i < ROW_V4; i += BLOCK) row4[i] = xr4[i];
  }
#endif
  __syncthreads();

  const float4* row4 = (const float4*)row;

  // ---------------- Fused max + sum over the row (b128 LDS reads) ----------
  float mx = -3.402823466e+38f;
  float ps = 0.0f;
  for (int i = tid; i < ROW_V4; i += BLOCK) {
    float4 v = row4[i];
    mx = fmaxf(fmaxf(mx, v.x), fmaxf(v.y, fmaxf(v.z, v.w)));
    ps += (v.x + v.y) + (v.z + v.w);
  }
  for (int off = 16; off > 0; off >>= 1) {
    mx  = fmaxf(mx, __shfl_xor(mx, off, 32));
    ps += __shfl_xor(ps, off, 32);
  }
  if (lane == 0) { redA[wave] = mx; redB[wave] = ps; }
  __syncthreads();
  mx = redA[0];
  ps = redB[0];
  for (int w = 1; w < NWAVE; ++w) {
    mx = fmaxf(mx, redA[w]);
    ps += redB[w];
  }
  __syncthreads();

  // Relative tau (tau of the max-shifted problem); tau in [-1, 0).
  float tau = (ps - (float)ROW_D * mx - 1.0f) / (float)ROW_D;

  // ---------------- Michelot fixed-point: tau = (sum_{v>tau} - 1)/k --------
  // tau is monotone non-decreasing => active set only shrinks => once the
  // count fits in CAP we can compact and never overflow.
  float kprev = (float)ROW_D;
  bool  compacted = false;
  int   ncand = 0;

  for (int it = 0; it < ITERS; ++it) {
    const float thresh = mx + tau;
    float s = 0.0f, k = 0.0f;

    if (!compacted && kprev <= (float)CAP) {
      // ---- compaction iteration: count+sum pass, block scan, write pass ---
      int c = 0;
      for (int i = tid; i < ROW_V4; i += BLOCK) {
        float4 v = row4[i];
        if (v.x > thresh) { s += v.x; ++c; }
        if (v.y > thresh) { s += v.y; ++c; }
        if (v.z > thresh) { s += v.z; ++c; }
        if (v.w > thresh) { s += v.w; ++c; }
      }
      // wave-level inclusive scan of counts
      int incl = c;
      for (int off = 1; off < 32; off <<= 1) {
        int y = __shfl_up(incl, off, 32);
        if (lane >= off) incl += y;
      }
      if (lane == 31) iredA[wave] = incl;
      __syncthreads();
      if (tid < NWAVE) {  // wave 0 scans the per-wave totals
        int w = iredA[tid];
        for (int off = 1; off < NWAVE; off <<= 1) {
          int y = __shfl_up(w, off, 32);
          if (lane >= off) w += y;
        }
        iredB[tid] = w;  // inclusive scan of wave totals
      }
      __syncthreads();
      int base = (wave > 0 ? iredB[wave - 1] : 0) + (incl - c);
      ncand = iredB[NWAVE - 1];
      // ordered write pass (deterministic: slots fixed by thread/index order)
      int pos = base;
      for (int i = tid; i < ROW_V4; i += BLOCK) {
        float4 v = row4[i];
        if (v.x > thresh) cand[pos++] = v.x;
        if (v.y > thresh) cand[pos++] = v.y;
        if (v.z > thresh) cand[pos++] = v.z;
        if (v.w > thresh) cand[pos++] = v.w;
      }
      // block-reduce s; k comes exactly from the scan total
      for (int off = 16; off > 0; off >>= 1) s += __shfl_xor(s, off, 32);
      if (lane == 0) redA[wave] = s;
      __syncthreads();
      s = redA[0];
      for (int w = 1; w < NWAVE; ++w) s += redA[w];
      __syncthreads();
      k = (float)ncand;
      compacted = true;
    } else if (compacted) {
      // ---- cheap iteration on the compacted active set --------------------
      for (int i = tid; i < ncand; i += BLOCK) {
        float v = cand[i];
        if (v > thresh) { s += v; k += 1.0f; }
      }
      for (int off = 16; off > 0; off >>= 1) {
        s += __shfl_xor(s, off, 32);
        k += __shfl_xor(k, off, 32);
      }
      if (lane == 0) { redA[wave] = s; redB[wave] = k; }
      __syncthreads();
      s = redA[0];
      k = redB[0];
      for (int w = 1; w < NWAVE; ++w) { s += redA[w]; k += redB[w]; }
      __syncthreads();
    } else {
      // ---- full-row iteration (b128 LDS reads) ----------------------------
      for (int i = tid; i < ROW_V4; i += BLOCK) {
        float4 v = row4[i];
        if (v.x > thresh) { s += v.x; k += 1.0f; }
        if (v.y > thresh) { s += v.y; k += 1.0f; }
        if (v.z > thresh) { s += v.z; k += 1.0f; }
        if (v.w > thresh) { s += v.w; k += 1.0f; }
      }
      for (int off = 16; off > 0; off >>= 1) {
        s += __shfl_xor(s, off, 32);
        k += __shfl_xor(k, off, 32);
      }
      if (lane == 0) { redA[wave] = s; redB[wave] = k; }
      __syncthreads();
      s = redA[0];
      k = redB[0];
      for (int w = 1; w < NWAVE; ++w) { s += redA[w]; k += redB[w]; }
      __syncthreads();
    }

    // k >= 1 always (the max element is above any tau < 0).
    tau = (s - k * mx - 1.0f) / k;
    kprev = k;
  }

  // ---------------- Output pass: out = relu(x - (mx + tau)), b128 ----------
  const float th = mx + tau;
  float4* out4 = (float4*)outr;
  for (int i = tid; i < ROW_V4; i += BLOCK) {
    float4 v = row4[i];
    float4 o;
    o.x = fmaxf(v.x - th, 0.0f);
    o.y = fmaxf(v.y - th, 0.0f);
    o.z = fmaxf(v.z - th, 0.0f);
    o.w = fmaxf(v.w - th, 0.0f);
    out4[i] = o;
  }
}

extern "C" void kernel_launch(void* const* d_in, const int* in_sizes, int n_in,
                              void* d_out, int out_size, void* d_ws, size_t ws_size,
                              hipStream_t stream) {
  (void)n_in; (void)d_ws; (void)ws_size; (void)out_size;
  const float* x = (const float*)d_in[0];
  float* out = (float*)d_out;

  const int rows = in_sizes[0] / ROW_D;  // 4096
  const size_t shmem =
      (size_t)(ROW_D + CAP + 2 * NWAVE) * sizeof(float) + 2 * NWAVE * sizeof(int);

  // Allow >64KB dynamic LDS (gfx1250 WGP has 320 KB).
  (void)hipFuncSetAttribute((const void*)sparsemax_row_kernel,
                            hipFuncAttributeMaxDynamicSharedMemorySize,
                            (int)shmem);

  sparsemax_row_kernel<<<rows, BLOCK, shmem, stream>>>(x, out);
}